// Critic_77171972374916
// MI455X (gfx1250) — compile-verified
//
#include <hip/hip_runtime.h>

#define N_NODES  50000
#define N_EDGES  800000
#define N_GRAPHS 512
#define D        64
#define NBLK     (N_NODES / 16)   // 3125, exact

typedef __attribute__((ext_vector_type(2))) float v2f;
typedef __attribute__((ext_vector_type(8))) float v8f;

// ---------------------------------------------------------------- copy: agg = x
__global__ void gin_copy_kernel(const float* __restrict__ src, float* __restrict__ dst, int n4) {
    int i = blockIdx.x * blockDim.x + threadIdx.x;
    if (i < n4) ((float4*)dst)[i] = ((const float4*)src)[i];
}

// ------------------------------------------- edge scatter: agg[dst] += x[src]
// 16 threads per edge, float4 each: coalesced B128 reads, 4 f32 atomics/lane.
__global__ void gin_scatter_kernel(const float* __restrict__ x,
                                   const int* __restrict__ src,
                                   const int* __restrict__ dst,
                                   float* __restrict__ agg) {
    int i = blockIdx.x * blockDim.x + threadIdx.x;
    if (i >= N_EDGES * 16) return;
    int e = i >> 4, q = i & 15;
    int s = src[e], d = dst[e];
    const float4 v = *(const float4*)(x + (size_t)s * D + q * 4);
    float* p = agg + (size_t)d * D + q * 4;
    atomicAdd(p + 0, v.x);
    atomicAdd(p + 1, v.y);
    atomicAdd(p + 2, v.z);
    atomicAdd(p + 3, v.w);
}

// -------------------------------- fused MLP: out = relu(relu(agg@wa+ba)@wb+bb)
// One wave per 16-node block; v_wmma_f32_16x16x4_f32 (exact fp32 WMMA).
__global__ void __launch_bounds__(256)
gin_mlp_kernel(const float* __restrict__ agg,
               const float* __restrict__ wa, const float* __restrict__ ba,
               const float* __restrict__ wb, const float* __restrict__ bb,
               float* __restrict__ out) {
    __shared__ float lds_h[8 * 16 * D];           // 8 waves x (16 x 64) = 32 KB
    const int wave = threadIdx.x >> 5;
    const int lane = threadIdx.x & 31;
    int blk = blockIdx.x * 8 + wave;
    if (blk >= NBLK) blk = NBLK - 1;              // duplicate last block; keeps EXEC/-barrier uniform
    const int row0 = blk * 16;
    const int lm = lane & 15;                     // A: row M   B/C/D: col N
    const int lh = lane >> 4;                     // lane-half selects K/M offset
    float* hbuf = lds_h + wave * (16 * D);

    // ---- GEMM1: h = relu(agg_tile @ wa + ba)
    v8f acc[4];
#pragma unroll
    for (int t = 0; t < 4; ++t) {
        float b = ba[t * 16 + lm];
        acc[t] = (v8f){b, b, b, b, b, b, b, b};
    }
#pragma unroll
    for (int k0 = 0; k0 < D; k0 += 4) {
        // A frag (16x4): lane holds row M=lm, K = k0 + 2*lh + v
        v2f a;
        const float* ap = agg + (size_t)(row0 + lm) * D + k0 + 2 * lh;
        a.x = ap[0]; a.y = ap[1];
#pragma unroll
        for (int t = 0; t < 4; ++t) {
            // B frag (4x16): lane holds col N=lm, K rows k0+2*lh and k0+2*lh+1
            v2f b;
            const float* bp = wa + (size_t)(k0 + 2 * lh) * D + t * 16 + lm;
            b.x = bp[0]; b.y = bp[D];
            acc[t] = __builtin_amdgcn_wmma_f32_16x16x4_f32(
                false, a, false, b, (short)0, acc[t], false, false);
        }
    }
    // relu + stage h to LDS row-major (C layout: M = v + 8*lh, N = t*16 + lm)
#pragma unroll
    for (int t = 0; t < 4; ++t)
#pragma unroll
        for (int v = 0; v < 8; ++v) {
            float val = acc[t][v];
            hbuf[(v + 8 * lh) * D + t * 16 + lm] = val > 0.f ? val : 0.f;
        }
    __syncthreads();

    // ---- GEMM2: out = relu(h @ wb + bb)
    v8f acc2[4];
#pragma unroll
    for (int t = 0; t < 4; ++t) {
        float b = bb[t * 16 + lm];
        acc2[t] = (v8f){b, b, b, b, b, b, b, b};
    }
#pragma unroll
    for (int k0 = 0; k0 < D; k0 += 4) {
        v2f a;
        const float* ap = hbuf + lm * D + k0 + 2 * lh;
        a.x = ap[0]; a.y = ap[1];
#pragma unroll
        for (int t = 0; t < 4; ++t) {
            v2f b;
            const float* bp = wb + (size_t)(k0 + 2 * lh) * D + t * 16 + lm;
            b.x = bp[0]; b.y = bp[D];
            acc2[t] = __builtin_amdgcn_wmma_f32_16x16x4_f32(
                false, a, false, b, (short)0, acc2[t], false, false);
        }
    }
#pragma unroll
    for (int t = 0; t < 4; ++t)
#pragma unroll
        for (int v = 0; v < 8; ++v) {
            float val = acc2[t][v];
            out[(size_t)(row0 + v + 8 * lh) * D + t * 16 + lm] = val > 0.f ? val : 0.f;
        }
}

// ---------------------------------------------------------------- pooling
__global__ void gin_zero_kernel(float* p, int n) {
    int i = blockIdx.x * blockDim.x + threadIdx.x;
    if (i < n) p[i] = 0.f;
}

__global__ void gin_pool_kernel(const float* __restrict__ x,
                                const int* __restrict__ batch,
                                float* __restrict__ pooled,
                                float* __restrict__ cnt) {
    int i = blockIdx.x * blockDim.x + threadIdx.x;
    if (i >= N_NODES * 16) return;
    int n = i >> 4, q = i & 15;
    int g = batch[n];
    const float4 v = *(const float4*)(x + (size_t)n * D + q * 4);
    float* p = pooled + (size_t)g * D + q * 4;
    atomicAdd(p + 0, v.x);
    atomicAdd(p + 1, v.y);
    atomicAdd(p + 2, v.z);
    atomicAdd(p + 3, v.w);
    if (q == 0) atomicAdd(cnt + g, 1.0f);
}

__global__ void gin_final_kernel(const float* __restrict__ pooled,
                                 const float* __restrict__ cnt,
                                 const float* __restrict__ wl,
                                 const float* __restrict__ bl,
                                 float* __restrict__ out) {
    int g = blockIdx.x * blockDim.x + threadIdx.x;
    if (g >= N_GRAPHS) return;
    float c = cnt[g];
    c = c > 1.f ? c : 1.f;
    float s = 0.f;
#pragma unroll
    for (int k = 0; k < D; ++k) s += pooled[g * D + k] * wl[k];
    out[g] = s / c + bl[0];
}

// ---------------------------------------------------------------- launcher
extern "C" void kernel_launch(void* const* d_in, const int* in_sizes, int n_in,
                              void* d_out, int out_size, void* d_ws, size_t ws_size,
                              hipStream_t stream) {
    (void)in_sizes; (void)n_in; (void)out_size; (void)ws_size;

    const float* features = (const float*)d_in[0];
    const int*   edge     = (const int*)d_in[1];
    const int*   batch    = (const int*)d_in[2];
    const float* w1a = (const float*)d_in[3];  const float* b1a = (const float*)d_in[4];
    const float* w1b = (const float*)d_in[5];  const float* b1b = (const float*)d_in[6];
    const float* w2a = (const float*)d_in[7];  const float* b2a = (const float*)d_in[8];
    const float* w2b = (const float*)d_in[9];  const float* b2b = (const float*)d_in[10];
    const float* w3a = (const float*)d_in[11]; const float* b3a = (const float*)d_in[12];
    const float* w3b = (const float*)d_in[13]; const float* b3b = (const float*)d_in[14];
    const float* wl  = (const float*)d_in[15]; const float* bl  = (const float*)d_in[16];

    const int* src = edge;             // edge_index[0, :]
    const int* dst = edge + N_EDGES;   // edge_index[1, :]

    float* xA     = (float*)d_ws;                      // N_NODES*D
    float* xB     = xA + (size_t)N_NODES * D;          // N_NODES*D
    float* agg    = xB + (size_t)N_NODES * D;          // N_NODES*D
    float* pooled = agg + (size_t)N_NODES * D;         // N_GRAPHS*D
    float* cnt    = pooled + (size_t)N_GRAPHS * D;     // N_GRAPHS (contiguous after pooled)

    const dim3 B(256);
    const int nd4        = N_NODES * D / 4;
    const dim3 gCopy((nd4 + 255) / 256);
    const dim3 gScat((N_EDGES * 16 + 255) / 256);
    const dim3 gMlp((NBLK + 7) / 8);

    // layer 1
    gin_copy_kernel<<<gCopy, B, 0, stream>>>(features, agg, nd4);
    gin_scatter_kernel<<<gScat, B, 0, stream>>>(features, src, dst, agg);
    gin_mlp_kernel<<<gMlp, B, 0, stream>>>(agg, w1a, b1a, w1b, b1b, xA);
    // layer 2
    gin_copy_kernel<<<gCopy, B, 0, stream>>>(xA, agg, nd4);
    gin_scatter_kernel<<<gScat, B, 0, stream>>>(xA, src, dst, agg);
    gin_mlp_kernel<<<gMlp, B, 0, stream>>>(agg, w2a, b2a, w2b, b2b, xB);
    // layer 3
    gin_copy_kernel<<<gCopy, B, 0, stream>>>(xB, agg, nd4);
    gin_scatter_kernel<<<gScat, B, 0, stream>>>(xB, src, dst, agg);
    gin_mlp_kernel<<<gMlp, B, 0, stream>>>(agg, w3a, b3a, w3b, b3b, xA);
    // pooling (pooled + cnt zeroed together; ws is poisoned, must re-init every call)
    gin_zero_kernel<<<(N_GRAPHS * D + N_GRAPHS + 255) / 256, B, 0, stream>>>(pooled, N_GRAPHS * D + N_GRAPHS);
    gin_pool_kernel<<<(N_NODES * 16 + 255) / 256, B, 0, stream>>>(xA, batch, pooled, cnt);
    gin_final_kernel<<<(N_GRAPHS + 255) / 256, B, 0, stream>>>(pooled, cnt, wl, bl, (float*)d_out);
}